// OSAR_31980326486393
// MI455X (gfx1250) — compile-verified
//
#include <hip/hip_runtime.h>
#include <hip/hip_bf16.h>

// Compressive-transformer memory shift for MI455X (gfx1250).
//
// Math reduction (see analysis): since SHORT_UNITS / RATE == B exactly, the
// compressed slices all come from the ORIGINAL memory, so
//   comp(128,1024) = memory[512:].view(128,4096) @ kernel.view(4096,1024) + bias
// and every output row of mems(128,1024,1024) is a plain copy of a row of
// {memory, comp, inputs}. Phase 1: f32 WMMA GEMM into workspace. Phase 2:
// row-assembly copy with nontemporal b128 stores (512 MiB store-bound,
// ~22us at 23.3 TB/s; all sources are L2-resident in the 192 MB L2).

typedef __attribute__((ext_vector_type(2))) float v2f;
typedef __attribute__((ext_vector_type(4))) float v4f;
typedef __attribute__((ext_vector_type(8))) float v8f;

#define FDIM 1024
#define KDIM 4096   // RATE * FDIM
#define CONVU 512

// ---------------------------------------------------------------------------
// Phase 1: comp = A(128x4096) @ B(4096x1024) + bias, f32 WMMA 16x16x4.
// One wave computes a 16(M) x 32(N) tile -> 8 x 32 = 256 waves total.
//
// VGPR layouts per CDNA5 ISA 7.12.2 (32-bit, wave32):
//   A 16x4 : lanes 0-15 -> M=lane, K={0,1} in v0/v1; lanes 16-31 -> K={2,3}
//   B 4x16 : lanes 0-15 -> N=lane, K={0,1} in v0/v1; lanes 16-31 -> K={2,3}
//   C 16x16: vgpr v -> M = v + (lane<16 ? 0 : 8), N = lane%16
// ---------------------------------------------------------------------------
__global__ void __launch_bounds__(128)
comp_gemm_wmma(const float* __restrict__ mem,   // (1024,1024) full memory
               const float* __restrict__ kern,  // (4,1024,1024) == (4096,1024)
               const float* __restrict__ bias,  // (1024,)
               float* __restrict__ comp)        // (128,1024) out
{
    const float* A = mem + CONVU * FDIM;        // (128, 4096) contiguous

    const int wid  = (blockIdx.x * blockDim.x + threadIdx.x) >> 5; // 0..255
    const int lane = threadIdx.x & 31;
    const int half = lane >> 4;   // 0: K={0,1}, 1: K={2,3}
    const int l    = lane & 15;

    const int m0 = (wid & 7) * 16;    // M tile base (0..112)
    const int c0 = (wid >> 3) * 32;   // N tile base (0..992)

    v8f acc0 = {};
    v8f acc1 = {};

    const float* arow  = A + (m0 + l) * KDIM + 2 * half; // contiguous K pair
    const float* bcol  = kern + c0 + l;                  // column l of B tile

    #pragma unroll 4
    for (int k = 0; k < KDIM; k += 4) {
        // A fragment: aligned float2 (K = k+2*half, k+2*half+1)
        v2f a = *(const v2f*)(arow + k);

        const float* bp = bcol + (k + 2 * half) * FDIM;
        v2f b0, b1;
        b0.x = bp[0];           // B[k+2h  ][c0+l]
        b0.y = bp[FDIM];        // B[k+2h+1][c0+l]
        b1.x = bp[16];          // B[k+2h  ][c0+16+l]
        b1.y = bp[FDIM + 16];   // B[k+2h+1][c0+16+l]

        acc0 = __builtin_amdgcn_wmma_f32_16x16x4_f32(
                   false, a, false, b0, (short)0, acc0, false, false);
        acc1 = __builtin_amdgcn_wmma_f32_16x16x4_f32(
                   false, a, false, b1, (short)0, acc1, false, false);
    }

    // Epilogue: N is constant per lane across the 8 C VGPRs -> one bias each.
    const int col0 = c0 + l;
    const int col1 = c0 + 16 + l;
    const float bb0 = bias[col0];
    const float bb1 = bias[col1];
    const int rbase = m0 + (half ? 8 : 0);

    #pragma unroll
    for (int v = 0; v < 8; ++v) {
        comp[(rbase + v) * FDIM + col0] = acc0[v] + bb0;
        comp[(rbase + v) * FDIM + col1] = acc1[v] + bb1;
    }
}

// ---------------------------------------------------------------------------
// Phase 2: assemble mems(128,1024,1024). One block per output row; uniform
// (scalar-branch) source classification, then a 4 KB row copy as float4 with
// nontemporal stores (512 MiB write stream should not thrash L2).
//   conv  rows r<512 : mem[b+1+r]            if r < 511-b, else comp[r-(511-b)]
//   short rows s=r-512: mem[512+4(b+1)+s]    if s < 512-4(b+1),
//                       else inputs row (s - (512-4(b+1)))
// ---------------------------------------------------------------------------
__global__ void __launch_bounds__(256)
assemble_rows(const float* __restrict__ mem,     // (1024,1024)
              const float* __restrict__ inputs,  // (128,4,1024) == (512,1024)
              const float* __restrict__ comp,    // (128,1024)
              float* __restrict__ out)           // (128,1024,1024)
{
    const int r = blockIdx.x;   // 0..1023 row within snapshot
    const int b = blockIdx.y;   // 0..127  snapshot index

    const float* src;
    if (r < CONVU) {
        const int shift = CONVU - 1 - b;            // 511-b
        src = (r < shift) ? (mem  + (b + 1 + r) * FDIM)
                          : (comp + (r - shift) * FDIM);
    } else {
        const int s    = r - CONVU;
        const int keep = CONVU - 4 * (b + 1);       // rows kept from memory
        src = (s < keep) ? (mem    + (CONVU + 4 * (b + 1) + s) * FDIM)
                         : (inputs + (s - keep) * FDIM);
    }

    const int t = threadIdx.x;                      // 0..255, 4 floats each
    v4f v = *((const v4f*)src + t);
    v4f* dst = (v4f*)(out + ((size_t)b * 1024 + r) * FDIM) + t;
    __builtin_nontemporal_store(v, dst);
}

extern "C" void kernel_launch(void* const* d_in, const int* in_sizes, int n_in,
                              void* d_out, int out_size, void* d_ws, size_t ws_size,
                              hipStream_t stream) {
    const float* inputs = (const float*)d_in[0];  // (128,4,1024)
    const float* memory = (const float*)d_in[1];  // (1024,1024)
    const float* kern   = (const float*)d_in[2];  // (4,1024,1024)
    const float* bias   = (const float*)d_in[3];  // (1024,)
    float* out  = (float*)d_out;                  // (128,1024,1024)
    float* comp = (float*)d_ws;                   // 128*1024 f32 = 512 KiB

    // Phase 1: 256 waves = 64 blocks x 128 threads (4 waves/block).
    comp_gemm_wmma<<<dim3(64), dim3(128), 0, stream>>>(memory, kern, bias, comp);

    // Phase 2: one 256-thread block per output row (1024 x 128 rows).
    assemble_rows<<<dim3(1024, 128), dim3(256), 0, stream>>>(memory, inputs, comp, out);
}